// NodeGCN_32856499815216
// MI455X (gfx1250) — compile-verified
//
#include <hip/hip_runtime.h>
#include <cstdint>

typedef __attribute__((ext_vector_type(2))) float v2f;
typedef __attribute__((ext_vector_type(8))) float v8f;

#define NFEAT 128
#define HID   20
#define NCLS  10
#define CATW  (3 * HID)   // 60

// ---------------------------------------------------------------------------
// Degree / normalization kernels
// ---------------------------------------------------------------------------
__global__ void deg_init_kernel(float* deg, int n) {
    int i = blockIdx.x * blockDim.x + threadIdx.x;
    if (i < n) deg[i] = 1.0f;  // self-loop weight
}

__global__ void deg_scatter_kernel(const int* __restrict__ col,
                                   const float* __restrict__ ew,
                                   float* deg, int e) {
    int i = blockIdx.x * blockDim.x + threadIdx.x;
    if (i < e) atomicAdd(&deg[col[i]], ew[i]);
}

__global__ void deg_finalize_kernel(float* dinv, int n) {
    int i = blockIdx.x * blockDim.x + threadIdx.x;
    if (i < n) {
        float d = dinv[i];
        dinv[i] = (d > 0.0f) ? rsqrtf(fmaxf(d, 1e-12f)) : 0.0f;
    }
}

// ---------------------------------------------------------------------------
// WMMA f32 GEMM:  out[:, col_ofs : col_ofs+NC] = A(MxK, lda) @ W(KxNC) [+bias]
//
// K, NC compile-time -> K-loop fully unrolls (K/4 v_wmma_f32_16x16x4_f32).
// W is staged into LDS transposed + zero-padded to the 16-wide N tile:
//   ldsW[c][k], pitch K+1 (odd pitch -> conflict-free per-lane ds_load_b64).
// Inner step: 1x global_load_b64 (A frag) + 1x ds_load_b64 (B frag) + 1 WMMA.
// No divergent branches in the loop -> EXEC all-ones around every WMMA.
//
// Fragments (per CDNA5 ISA layouts, lo=lane%16, hi=lane/16):
//   A 16x4 f32 : vgpr0 = A[row0+lo][k0+2hi],  vgpr1 = A[row0+lo][k0+2hi+1]
//   B 4x16 f32 : vgpr0 = W[k0+2hi][n0+lo],    vgpr1 = W[k0+2hi+1][n0+lo]
//   C/D 16x16  : vgpr v, lane: row = row0 + v + 8*hi, col = n0+lo
// ---------------------------------------------------------------------------
template <int K, int NC>
__global__ void gemm_wmma_f32(const float* __restrict__ A, int lda,
                              const float* __restrict__ W,
                              const float* __restrict__ bias,  // may be null
                              float* __restrict__ out, int ldo, int col_ofs,
                              int M) {
    constexpr int NTILES = (NC + 15) / 16;
    constexpr int PADC   = NTILES * 16;   // padded column count
    constexpr int PITCH  = K + 1;         // odd pitch -> distinct LDS banks per lane

    __shared__ float ldsW[PADC][PITCH];

    // Cooperative stage of W^T (zero-padded columns), once per block.
    for (int idx = threadIdx.x; idx < PADC * K; idx += blockDim.x) {
        const int c = idx / K;
        const int k = idx - c * K;
        ldsW[c][k] = (c < NC) ? W[(size_t)k * NC + c] : 0.0f;
    }
    __syncthreads();

    const int lane = threadIdx.x & 31;
    const int wave = threadIdx.x >> 5;
    const int waves_per_blk = blockDim.x >> 5;

    const int mtiles = (M + 15) >> 4;
    const int tile = blockIdx.x * waves_per_blk + wave;  // wave-uniform
    if (tile >= mtiles * NTILES) return;                 // uniform branch

    const int mt = tile / NTILES;
    const int nt = tile - mt * NTILES;
    const int row0 = mt << 4;
    const int n0 = nt << 4;

    const int lo = lane & 15;
    const int hi = lane >> 4;
    const int n = n0 + lo;                 // this lane's output column

    v8f c = {};
    if (bias != nullptr) {
        const float bv = (n < NC) ? bias[n] : 0.0f;
        #pragma unroll
        for (int i = 0; i < 8; ++i) c[i] = bv;
    }

    const int arow = (row0 + lo < M) ? (row0 + lo) : (M - 1);
    const float* __restrict__ aptr = A + (size_t)arow * lda + 2 * hi;
    const float* __restrict__ bptr = &ldsW[n][2 * hi];

    #pragma unroll
    for (int k0 = 0; k0 < K; k0 += 4) {
        v2f a, b;
        a.x = aptr[k0];          // global_load_b64, immediate offset
        a.y = aptr[k0 + 1];
        b.x = bptr[k0];          // ds_load_b64, immediate offset
        b.y = bptr[k0 + 1];
        c = __builtin_amdgcn_wmma_f32_16x16x4_f32(false, a, false, b,
                                                  (short)0, c, false, false);
    }

    if (n < NC) {
        #pragma unroll
        for (int v = 0; v < 8; ++v) {
            const int r = row0 + v + 8 * hi;
            if (r < M) out[(size_t)r * ldo + col_ofs + n] = c[v];
        }
    }
}

// ---------------------------------------------------------------------------
// Self-loop contribution: cat_blk[i][f] = h[i][f] * dinv[i]^2   (float4 path)
// One thread per (node, quad-of-4-features): HID/4 = 5 quads.
// ---------------------------------------------------------------------------
__global__ void agg_init_kernel(const float* __restrict__ h,
                                const float* __restrict__ dinv,
                                float* __restrict__ cat_blk,
                                int n) {
    int idx = blockIdx.x * blockDim.x + threadIdx.x;
    if (idx >= n * (HID / 4)) return;
    const int node = idx / (HID / 4);
    const int q = idx - node * (HID / 4);
    const float di = dinv[node];
    const float s = di * di;
    float4 v = ((const float4*)(h + (size_t)node * HID))[q];
    v.x *= s; v.y *= s; v.z *= s; v.w *= s;
    ((float4*)(cat_blk + (size_t)node * CATW))[q] = v;
}

// ---------------------------------------------------------------------------
// Edge scatter: agg[col] += h[row] * (dinv[row]*ew*dinv[col])
// b128 row gathers (L2-resident), global_atomic_add_f32 scatter.
// ---------------------------------------------------------------------------
__global__ void edge_scatter_kernel(const int* __restrict__ row,
                                    const int* __restrict__ col,
                                    const float* __restrict__ ew,
                                    const float* __restrict__ h,
                                    const float* __restrict__ dinv,
                                    float* __restrict__ cat_blk,
                                    int e) {
    int i = blockIdx.x * blockDim.x + threadIdx.x;
    if (i >= e) return;
    const int r = row[i];
    const int c = col[i];
    const float norm = dinv[r] * ew[i] * dinv[c];

    const float4* __restrict__ hr = (const float4*)(h + (size_t)r * HID);
    float* __restrict__ ac = cat_blk + (size_t)c * CATW;

    #pragma unroll
    for (int q = 0; q < HID / 4; ++q) {
        float4 v = hr[q];
        atomicAdd(&ac[4 * q + 0], v.x * norm);
        atomicAdd(&ac[4 * q + 1], v.y * norm);
        atomicAdd(&ac[4 * q + 2], v.z * norm);
        atomicAdd(&ac[4 * q + 3], v.w * norm);
    }
}

// ---------------------------------------------------------------------------
// Post: v = agg + b; v /= max(||v||2, 1e-12); relu; in place (float4 path)
// ---------------------------------------------------------------------------
__global__ void post_kernel(float* __restrict__ cat_blk,
                            const float* __restrict__ bias, int n) {
    int i = blockIdx.x * blockDim.x + threadIdx.x;
    if (i >= n) return;
    float4* p = (float4*)(cat_blk + (size_t)i * CATW);
    const float4* b4 = (const float4*)bias;

    float4 v[HID / 4];
    float ss = 0.0f;
    #pragma unroll
    for (int q = 0; q < HID / 4; ++q) {
        float4 t = p[q];
        float4 bb = b4[q];
        t.x += bb.x; t.y += bb.y; t.z += bb.z; t.w += bb.w;
        ss += t.x * t.x + t.y * t.y + t.z * t.z + t.w * t.w;
        v[q] = t;
    }
    const float inv = 1.0f / fmaxf(sqrtf(ss), 1e-12f);
    #pragma unroll
    for (int q = 0; q < HID / 4; ++q) {
        float4 t = v[q];
        t.x = fmaxf(t.x * inv, 0.0f);
        t.y = fmaxf(t.y * inv, 0.0f);
        t.z = fmaxf(t.z * inv, 0.0f);
        t.w = fmaxf(t.w * inv, 0.0f);
        p[q] = t;
    }
}

// ---------------------------------------------------------------------------
// Launch
// ---------------------------------------------------------------------------
extern "C" void kernel_launch(void* const* d_in, const int* in_sizes, int n_in,
                              void* d_out, int out_size, void* d_ws, size_t ws_size,
                              hipStream_t stream) {
    const float* x  = (const float*)d_in[0];
    const int*   ei = (const int*)d_in[1];
    const float* ew = (const float*)d_in[2];
    const float* Wc[3] = {(const float*)d_in[3], (const float*)d_in[5], (const float*)d_in[7]};
    const float* bc[3] = {(const float*)d_in[4], (const float*)d_in[6], (const float*)d_in[8]};
    const float* Wl = (const float*)d_in[9];
    const float* bl = (const float*)d_in[10];
    float* out = (float*)d_out;

    const int N = in_sizes[0] / NFEAT;   // 100000
    const int E = in_sizes[1] / 2;       // 3200000
    const int* row = ei;
    const int* col = ei + E;

    // Workspace layout (floats), 1KB-aligned blocks
    float* ws   = (float*)d_ws;
    size_t off1 = ((size_t)N + 255) & ~(size_t)255;
    size_t off2 = off1 + (((size_t)N * HID + 255) & ~(size_t)255);
    float* dinv = ws;                    // N
    float* h    = ws + off1;             // N * HID
    float* cat  = ws + off2;             // N * 60

    const int TB = 256;
    const int waves_per_blk = TB / 32;
    dim3 blk(TB);

    // Degree + normalization
    deg_init_kernel<<<dim3((N + TB - 1) / TB), blk, 0, stream>>>(dinv, N);
    deg_scatter_kernel<<<dim3((E + TB - 1) / TB), blk, 0, stream>>>(col, ew, dinv, E);
    deg_finalize_kernel<<<dim3((N + TB - 1) / TB), blk, 0, stream>>>(dinv, N);

    const int mtiles = (N + 15) / 16;

    for (int l = 0; l < 3; ++l) {
        // h = in @ W_l   (WMMA f32, fully unrolled K loop)
        if (l == 0) {
            int tiles = mtiles * ((HID + 15) / 16);
            int blocks = (tiles + waves_per_blk - 1) / waves_per_blk;
            gemm_wmma_f32<NFEAT, HID><<<dim3(blocks), blk, 0, stream>>>(
                x, NFEAT, Wc[0], nullptr, h, HID, 0, N);
        } else {
            int tiles = mtiles * ((HID + 15) / 16);
            int blocks = (tiles + waves_per_blk - 1) / waves_per_blk;
            gemm_wmma_f32<HID, HID><<<dim3(blocks), blk, 0, stream>>>(
                cat + (l - 1) * HID, CATW, Wc[l], nullptr, h, HID, 0, N);
        }

        float* cat_blk = cat + l * HID;
        agg_init_kernel<<<dim3((N * (HID / 4) + TB - 1) / TB), blk, 0, stream>>>(
            h, dinv, cat_blk, N);
        edge_scatter_kernel<<<dim3((E + TB - 1) / TB), blk, 0, stream>>>(
            row, col, ew, h, dinv, cat_blk, E);
        post_kernel<<<dim3((N + TB - 1) / TB), blk, 0, stream>>>(cat_blk, bc[l], N);
    }

    // Final linear: out = cat @ Wl + bl  (K=60, NC=10 padded to one 16-tile)
    {
        int tiles = mtiles * ((NCLS + 15) / 16);
        int blocks = (tiles + waves_per_blk - 1) / waves_per_blk;
        gemm_wmma_f32<CATW, NCLS><<<dim3(blocks), blk, 0, stream>>>(
            cat, CATW, Wl, bl, out, NCLS, 0, N);
    }
}